// LengthRegulator_51161650430547
// MI455X (gfx1250) — compile-verified
//
#include <hip/hip_runtime.h>

typedef __attribute__((ext_vector_type(2))) float v2f;
typedef __attribute__((ext_vector_type(4))) float f4;
typedef __attribute__((ext_vector_type(8))) float v8f;

constexpr int kB = 16;
constexpr int kT = 512;
constexpr int kD = 512;
constexpr int kL = 4096;   // mel_max_length

// ---------------------------------------------------------------------------
// Kernel 1: per-batch inclusive cumsum of durations (Hillis-Steele in LDS)
// grid = (kB), block = (kT)
// ---------------------------------------------------------------------------
__global__ __launch_bounds__(kT) void cumsum_kernel(const int* __restrict__ target,
                                                    int* __restrict__ cs) {
    __shared__ int s[kT];
    const int t = threadIdx.x;
    const int b = blockIdx.x;
    s[t] = target[b * kT + t];
    __syncthreads();
    #pragma unroll
    for (int off = 1; off < kT; off <<= 1) {
        int v = (t >= off) ? s[t - off] : 0;
        __syncthreads();
        s[t] += v;
        __syncthreads();
    }
    cs[b * kT + t] = s[t];
}

// ---------------------------------------------------------------------------
// Kernel 2: fused DurationPredictor
//   H = relu(X*W1 + b1)  (never leaves registers)
//   dp = relu(H*W2 + b2)
// grid = (kB*kT/16) blocks, 256 threads = 8 waves.
// Each block owns 16 rows of X (staged in 32KB LDS); each wave owns 4 N-tiles
// of 16 columns and drives v_wmma_f32_16x16x4_f32 over K=512.
// ---------------------------------------------------------------------------
__global__ __launch_bounds__(256) void dp_kernel(const float* __restrict__ x,
                                                 const float* __restrict__ W1,
                                                 const float* __restrict__ b1,
                                                 const float* __restrict__ W2,
                                                 const float* __restrict__ b2,
                                                 float* __restrict__ dp_out) {
    __shared__ float xs[16 * kD];    // 32 KB row block of x
    __shared__ float dp_lds[16];     // cross-wave dp accumulator

    const int tid      = threadIdx.x;
    const int lane     = tid & 31;
    const int wave     = tid >> 5;
    const int row_base = blockIdx.x * 16;

    // Stage the 16-row block of x into LDS (coalesced).
    for (int i = tid; i < 16 * kD; i += 256)
        xs[i] = x[row_base * kD + i];
    if (tid < 16) dp_lds[tid] = 0.0f;
    __syncthreads();

    const int m     = lane & 15;   // A row / B,C column within tile
    const int khalf = lane >> 4;   // K-pair selector: lanes 16-31 hold K+2,K+3

    float part[8];
    #pragma unroll
    for (int j = 0; j < 8; ++j) part[j] = 0.0f;

    // 32 N-tiles total / 8 waves = 4 tiles per wave
    for (int t4 = 0; t4 < 4; ++t4) {
        const int n0  = (wave * 4 + t4) * 16;
        const float b1v = b1[n0 + m];
        const float w2v = W2[n0 + m];

        v8f acc = {0.f, 0.f, 0.f, 0.f, 0.f, 0.f, 0.f, 0.f};

        for (int k0 = 0; k0 < kD; k0 += 4) {
            // A fragment 16x4: lane m holds X[row m][k0+2*khalf .. +1] (contiguous pair)
            const v2f a = *(const v2f*)&xs[m * kD + k0 + 2 * khalf];
            // B fragment 4x16: lane m holds W1[k0+2*khalf][n0+m], W1[k0+2*khalf+1][n0+m]
            v2f bm;
            bm.x = W1[(k0 + 2 * khalf)     * kD + n0 + m];
            bm.y = W1[(k0 + 2 * khalf + 1) * kD + n0 + m];
            acc = __builtin_amdgcn_wmma_f32_16x16x4_f32(
                false, a, false, bm, (short)0, acc, false, false);
        }

        // Fused epilogue: relu(+b1), scale by W2 column slice, accumulate.
        // C layout: VGPR j, lanes 0-15 -> row j (N=lane); lanes 16-31 -> row j+8.
        #pragma unroll
        for (int j = 0; j < 8; ++j) {
            float h = acc[j] + b1v;
            h = h > 0.0f ? h : 0.0f;
            part[j] += h * w2v;
        }
    }

    // Reduce over the 16 lanes of each half-wave (each half covers 16 N values).
    #pragma unroll
    for (int j = 0; j < 8; ++j) {
        float v = part[j];
        v += __shfl_xor(v, 1);
        v += __shfl_xor(v, 2);
        v += __shfl_xor(v, 4);
        v += __shfl_xor(v, 8);
        part[j] = v;
    }
    if ((lane & 15) == 0) {
        #pragma unroll
        for (int j = 0; j < 8; ++j)
            atomicAdd(&dp_lds[j + 8 * khalf], part[j]);
    }
    __syncthreads();

    if (tid < 16) {
        float v = dp_lds[tid] + b2[0];
        dp_out[row_base + tid] = v > 0.0f ? v : 0.0f;
    }
}

// ---------------------------------------------------------------------------
// Kernel 3: length regulator (cumsum + searchsorted gather, zero pad)
// grid = (kL, kB), block = 128. One block per output frame; each lane moves
// 16 bytes of the 2KB row. Non-temporal stores: 128MB write-once stream.
// ---------------------------------------------------------------------------
__global__ __launch_bounds__(128) void lr_kernel(const float* __restrict__ x,
                                                 const int* __restrict__ cs,
                                                 float* __restrict__ out) {
    __shared__ int s_cs[kT];
    const int t   = threadIdx.x;
    const int pos = blockIdx.x;
    const int b   = blockIdx.y;

    for (int i = t; i < kT; i += 128)
        s_cs[i] = cs[b * kT + i];
    __syncthreads();

    const int total = s_cs[kT - 1];

    f4 val = {0.f, 0.f, 0.f, 0.f};
    if (pos < total) {
        // upper_bound: first idx with cs[idx] > pos  (searchsorted side='right')
        int lo = 0, hi = kT;
        while (lo < hi) {
            int mid = (lo + hi) >> 1;
            if (s_cs[mid] <= pos) lo = mid + 1; else hi = mid;
        }
        const f4* src = (const f4*)(x + ((size_t)(b * kT + lo)) * kD);
        val = src[t];
    }
    f4* dst = (f4*)(out + ((size_t)(b * kL + pos)) * kD);
    __builtin_nontemporal_store(val, &dst[t]);
}

// ---------------------------------------------------------------------------
extern "C" void kernel_launch(void* const* d_in, const int* in_sizes, int n_in,
                              void* d_out, int out_size, void* d_ws, size_t ws_size,
                              hipStream_t stream) {
    const float* x      = (const float*)d_in[0];
    const float* W1     = (const float*)d_in[1];
    const float* b1     = (const float*)d_in[2];
    const float* W2     = (const float*)d_in[3];
    const float* b2     = (const float*)d_in[4];
    const int*   target = (const int*)d_in[5];
    (void)in_sizes; (void)n_in; (void)out_size; (void)ws_size;

    float* out = (float*)d_out;                       // [B, L, D]
    float* dp  = out + (size_t)kB * kL * kD;          // [B, T] appended
    int*   cs  = (int*)d_ws;                          // [B, T] scratch cumsum

    cumsum_kernel<<<kB, kT, 0, stream>>>(target, cs);
    dp_kernel<<<(kB * kT) / 16, 256, 0, stream>>>(x, W1, b1, W2, b2, dp);
    lr_kernel<<<dim3(kL, kB), 128, 0, stream>>>(x, cs, out);
}